// BidirectionalCrossAttentionGate_20658792694505
// MI455X (gfx1250) — compile-verified
//
#include <hip/hip_runtime.h>
#include <math.h>

#define NROWS 8192
#define DF 64
#define NFELEM (NROWS * DF)

typedef __attribute__((ext_vector_type(2))) float v2f;
typedef __attribute__((ext_vector_type(8))) float v8f;

// D(16x16,f32) = A(16x4,f32) * B(4x16,f32) + C    -> v_wmma_f32_16x16x4_f32
__device__ __forceinline__ v8f wmma4(v2f a, v2f b, v8f c) {
  return __builtin_amdgcn_wmma_f32_16x16x4_f32(false, a, false, b, (short)0, c,
                                               false, false);
}

__device__ __forceinline__ float sigm(float x) { return 1.0f / (1.0f + __expf(-x)); }

// ---------------------------------------------------------------------------
// Kernel 1: six 64x64 projections. out = (X @ W^T + b) * scale  (scale=1/8 for Q)
// one wave per 16-row tile; A/B layouts: lane&15 = M(or N), K = 4j + 2*hl + elem
// ---------------------------------------------------------------------------
__global__ void __launch_bounds__(32) proj_kernel(
    const float* __restrict__ self_x, const float* __restrict__ conv_x,
    const float* __restrict__ w0, const float* __restrict__ b0,
    const float* __restrict__ w1, const float* __restrict__ b1,
    const float* __restrict__ w2, const float* __restrict__ b2,
    const float* __restrict__ w3, const float* __restrict__ b3,
    const float* __restrict__ w4, const float* __restrict__ b4,
    const float* __restrict__ w5, const float* __restrict__ b5,
    float* __restrict__ ws) {
  int y = blockIdx.y;
  const float *X, *W, *B;
  float scale = 1.0f;
  switch (y) {
    case 0: X = self_x; W = w0; B = b0; scale = 0.125f; break;  // q1(self)
    case 1: X = conv_x; W = w1; B = b1; break;                  // k1(conv)
    case 2: X = conv_x; W = w2; B = b2; break;                  // v1(conv)
    case 3: X = conv_x; W = w3; B = b3; scale = 0.125f; break;  // q2(conv)
    case 4: X = self_x; W = w4; B = b4; break;                  // k2(self)
    default: X = self_x; W = w5; B = b5; break;                 // v2(self)
  }
  float* out = ws + (size_t)y * NFELEM;

  int lane = threadIdx.x & 31, l16 = lane & 15, hl = lane >> 4;
  int rowb = blockIdx.x * 16;

  v2f a[16];
#pragma unroll
  for (int j = 0; j < 16; ++j)
    a[j] = *(const v2f*)(X + (size_t)(rowb + l16) * DF + 4 * j + 2 * hl);

#pragma unroll
  for (int c = 0; c < 4; ++c) {
    float bb = B[16 * c + l16];
    v8f acc = {bb, bb, bb, bb, bb, bb, bb, bb};
#pragma unroll
    for (int j = 0; j < 16; ++j) {
      v2f b2 = *(const v2f*)(W + (size_t)(16 * c + l16) * DF + 4 * j + 2 * hl);
      acc = wmma4(a[j], b2, acc);
    }
#pragma unroll
    for (int r = 0; r < 8; ++r)
      out[(size_t)(rowb + r + 8 * hl) * DF + 16 * c + l16] = acc[r] * scale;
  }
}

// ---------------------------------------------------------------------------
// Kernel 2: flash attention, O = softmax(Q K^T) V   (1/sqrt(d) folded into Q)
// grid (N/64, 2). 4 waves/block, 16 q-rows per wave, KV super-tile of 64 rows
// staged in LDS (K row-major stride 68, V transposed stride 66).
// ---------------------------------------------------------------------------
__global__ void __launch_bounds__(128) flash_kernel(float* __restrict__ ws) {
  __shared__ float sK[64 * 68];    // K tile, [row][col], padded
  __shared__ float sVt[64 * 66];   // V tile transposed, [col][row], padded
  __shared__ float sP[4 * 16 * 18];  // per-wave P transpose buffer

  int dir = blockIdx.y;
  const float* Q = ws + (size_t)(dir ? 3 : 0) * NFELEM;
  const float* K = ws + (size_t)(dir ? 4 : 1) * NFELEM;
  const float* V = ws + (size_t)(dir ? 5 : 2) * NFELEM;
  float* O = ws + (size_t)(6 + dir) * NFELEM;

  int tid = threadIdx.x;
  int lane = tid & 31, l16 = lane & 15, hl = (lane >> 4) & 1;
  int wid = tid >> 5;
  int qb = blockIdx.x * 64 + wid * 16;
  float* Pw = sP + wid * (16 * 18);

  v2f aQ[16];
#pragma unroll
  for (int j = 0; j < 16; ++j)
    aQ[j] = *(const v2f*)(Q + (size_t)(qb + l16) * DF + 4 * j + 2 * hl);

  v8f oacc[4];
  float mrow[8], lsum[8], p[8];
#pragma unroll
  for (int c = 0; c < 4; ++c)
#pragma unroll
    for (int r = 0; r < 8; ++r) oacc[c][r] = 0.0f;
#pragma unroll
  for (int r = 0; r < 8; ++r) { mrow[r] = -3.0e38f; lsum[r] = 0.0f; }

  for (int t = 0; t < NROWS / 64; ++t) {
    __syncthreads();
    // cooperative stage of K (row-major) and V (transposed) into LDS
#pragma unroll
    for (int it = 0; it < 8; ++it) {
      int e = it * 128 + tid;
      int row = e >> 4, c4 = (e & 15) * 4;
      float4 k4 = *(const float4*)(K + (size_t)(t * 64 + row) * DF + c4);
      *(float4*)(sK + row * 68 + c4) = k4;
      float4 v4 = *(const float4*)(V + (size_t)(t * 64 + row) * DF + c4);
      sVt[(c4 + 0) * 66 + row] = v4.x;
      sVt[(c4 + 1) * 66 + row] = v4.y;
      sVt[(c4 + 2) * 66 + row] = v4.z;
      sVt[(c4 + 3) * 66 + row] = v4.w;
    }
    __syncthreads();

#pragma unroll
    for (int kb = 0; kb < 4; ++kb) {
      // S = Q * K^T for this 16x16 tile (C layout: row = r+8*hl, col = l16)
      v8f s;
#pragma unroll
      for (int r = 0; r < 8; ++r) s[r] = 0.0f;
      const float* kbase = sK + (kb * 16 + l16) * 68 + 2 * hl;
#pragma unroll
      for (int j = 0; j < 16; ++j) {
        v2f b2 = *(const v2f*)(kbase + 4 * j);
        s = wmma4(aQ[j], b2, s);
      }
      // online softmax per row (reduce over the 16 lanes of each half)
#pragma unroll
      for (int r = 0; r < 8; ++r) {
        float sv = s[r];
        float mx = sv;
        mx = fmaxf(mx, __shfl_xor(mx, 8));
        mx = fmaxf(mx, __shfl_xor(mx, 4));
        mx = fmaxf(mx, __shfl_xor(mx, 2));
        mx = fmaxf(mx, __shfl_xor(mx, 1));
        float mnew = fmaxf(mrow[r], mx);
        float corr = __expf(mrow[r] - mnew);
        float pv = __expf(sv - mnew);
        float ps = pv;
        ps += __shfl_xor(ps, 8);
        ps += __shfl_xor(ps, 4);
        ps += __shfl_xor(ps, 2);
        ps += __shfl_xor(ps, 1);
        lsum[r] = lsum[r] * corr + ps;
        mrow[r] = mnew;
        p[r] = pv;
        oacc[0][r] *= corr; oacc[1][r] *= corr;
        oacc[2][r] *= corr; oacc[3][r] *= corr;
      }
      // C-layout -> A-layout transpose of P through per-wave LDS buffer
#pragma unroll
      for (int r = 0; r < 8; ++r) Pw[(r + 8 * hl) * 18 + l16] = p[r];
      asm volatile("s_wait_dscnt 0" ::: "memory");
      // O += P * V
#pragma unroll
      for (int c = 0; c < 4; ++c) {
        const float* vb = sVt + (16 * c + l16) * 66 + kb * 16 + 2 * hl;
        const float* pb = Pw + l16 * 18 + 2 * hl;
#pragma unroll
        for (int jj = 0; jj < 4; ++jj) {
          v2f a2 = *(const v2f*)(pb + 4 * jj);
          v2f b2 = *(const v2f*)(vb + 4 * jj);
          oacc[c] = wmma4(a2, b2, oacc[c]);
        }
      }
    }
  }
#pragma unroll
  for (int c = 0; c < 4; ++c)
#pragma unroll
    for (int r = 0; r < 8; ++r)
      O[(size_t)(qb + r + 8 * hl) * DF + 16 * c + l16] = oacc[c][r] / lsum[r];
}

// ---------------------------------------------------------------------------
// Kernel 3: residual, pooling, tensor fusion (WMMA over K=4096), gate, fusion,
// final sigmoid. One wave per 16 rows.
// ---------------------------------------------------------------------------
__global__ void __launch_bounds__(32) tail_kernel(
    const float* __restrict__ o1g, const float* __restrict__ o2g,
    const float* __restrict__ p1w, const float* __restrict__ p1b,
    const float* __restrict__ p2w, const float* __restrict__ p2b,
    const float* __restrict__ tw, const float* __restrict__ tb,
    const float* __restrict__ gw, const float* __restrict__ gb,
    const float* __restrict__ fw, const float* __restrict__ fb,
    const float* __restrict__ rw, const float* __restrict__ rb,
    const float* __restrict__ ow, const float* __restrict__ ob,
    float* __restrict__ out) {
  __shared__ float sA[16 * 66];   // attn_out1 tile
  __shared__ float sB[16 * 66];   // attn_out2 tile
  __shared__ float sG1[16 * 66];  // gi1 tile
  __shared__ float sG2[16 * 66];  // gi2 tile
  __shared__ float sT[16 * 66];   // gelu(tensor) tile
  __shared__ float sH[16 * 196];  // gate * combined
  __shared__ float sStats[64];    // mean1,max1,mean2,max2 per row

  int tid = threadIdx.x, lane = tid & 31, l16 = lane & 15, hl = lane >> 4;
  int rowb = blockIdx.x * 16;

#pragma unroll
  for (int i = 0; i < 32; ++i) {
    int e = i * 32 + tid, r = e >> 6, col = e & 63;
    sA[r * 66 + col] = o1g[(size_t)(rowb + r) * 64 + col];
    sB[r * 66 + col] = o2g[(size_t)(rowb + r) * 64 + col];
  }
  __syncthreads();

  // residual = concat(o1,o2) @ rw^T + rb   (K = 128)
  v8f racc[4];
#pragma unroll
  for (int c = 0; c < 4; ++c) {
    float bb = rb[16 * c + l16];
#pragma unroll
    for (int r = 0; r < 8; ++r) racc[c][r] = bb;
  }
#pragma unroll
  for (int j = 0; j < 32; ++j) {
    int kk = 4 * j + 2 * hl;
    v2f a = (kk < 64) ? *(const v2f*)(sA + l16 * 66 + kk)
                      : *(const v2f*)(sB + l16 * 66 + (kk - 64));
#pragma unroll
    for (int c = 0; c < 4; ++c) {
      v2f b2 = *(const v2f*)(rw + (size_t)(16 * c + l16) * 128 + kk);
      racc[c] = wmma4(a, b2, racc[c]);
    }
  }

  // per-row mean/max pooling (lanes 0-15: o1 rows, lanes 16-31: o2 rows)
  {
    const float* src = hl ? (sB + l16 * 66) : (sA + l16 * 66);
    float sum = 0.0f, mx = -3.0e38f;
    for (int c2 = 0; c2 < 64; ++c2) { float v = src[c2]; sum += v; mx = fmaxf(mx, v); }
    sStats[l16 * 4 + hl * 2 + 0] = sum * (1.0f / 64.0f);
    sStats[l16 * 4 + hl * 2 + 1] = mx;
  }
  __syncthreads();

#pragma unroll
  for (int i = 0; i < 32; ++i) {
    int e = i * 32 + tid, r = e >> 6, col = e & 63;
    sG1[r * 66 + col] = p1w[col * 2] * sStats[r * 4 + 0] +
                        p1w[col * 2 + 1] * sStats[r * 4 + 1] + p1b[col];
    sG2[r * 66 + col] = p2w[col * 2] * sStats[r * 4 + 2] +
                        p2w[col * 2 + 1] * sStats[r * 4 + 3] + p2b[col];
  }
  __syncthreads();

  // tensor[n,k] = sum_ij gi1[n,i] gi2[n,j] W[k, i*64+j]  -> GEMM with K=4096
  v8f tacc[4];
#pragma unroll
  for (int c = 0; c < 4; ++c) {
    float bb = tb[16 * c + l16];
#pragma unroll
    for (int r = 0; r < 8; ++r) tacc[c][r] = bb;
  }
  for (int i = 0; i < 64; ++i) {
    float ga = sG1[l16 * 66 + i];
    const float* g2p = sG2 + l16 * 66 + 2 * hl;
#pragma unroll
    for (int jj = 0; jj < 16; ++jj) {
      v2f g2 = *(const v2f*)(g2p + 4 * jj);
      v2f a; a.x = ga * g2.x; a.y = ga * g2.y;
      int kk = 64 * i + 4 * jj + 2 * hl;
#pragma unroll
      for (int c = 0; c < 4; ++c) {
        v2f b2 = *(const v2f*)(tw + (size_t)(16 * c + l16) * 4096 + kk);
        tacc[c] = wmma4(a, b2, tacc[c]);
      }
    }
  }
  // exact-erf GELU, store in C layout
#pragma unroll
  for (int c = 0; c < 4; ++c)
#pragma unroll
    for (int r = 0; r < 8; ++r) {
      float x = tacc[c][r];
      sT[(r + 8 * hl) * 66 + 16 * c + l16] =
          0.5f * x * (1.0f + erff(x * 0.70710678118654752f));
    }
  __syncthreads();

  // gate = sigmoid(combined @ gw^T + gb), h = gate*combined (3 passes of 64 cols)
#pragma unroll
  for (int pass = 0; pass < 3; ++pass) {
    v8f gacc[4];
#pragma unroll
    for (int c = 0; c < 4; ++c) {
      float bb = gb[64 * pass + 16 * c + l16];
#pragma unroll
      for (int r = 0; r < 8; ++r) gacc[c][r] = bb;
    }
#pragma unroll
    for (int j = 0; j < 48; ++j) {
      int kk = 4 * j + 2 * hl;
      int seg = kk >> 6, off = kk & 63;
      const float* ap = (seg == 0) ? sG1 : ((seg == 1) ? sG2 : sT);
      v2f a = *(const v2f*)(ap + l16 * 66 + off);
#pragma unroll
      for (int c = 0; c < 4; ++c) {
        v2f b2 = *(const v2f*)(gw + (size_t)(64 * pass + 16 * c + l16) * 192 + kk);
        gacc[c] = wmma4(a, b2, gacc[c]);
      }
    }
    const float* cp = (pass == 0) ? sG1 : ((pass == 1) ? sG2 : sT);
#pragma unroll
    for (int c = 0; c < 4; ++c)
#pragma unroll
      for (int r = 0; r < 8; ++r) {
        int row = r + 8 * hl, col16 = 16 * c + l16;
        sH[row * 196 + 64 * pass + col16] = sigm(gacc[c][r]) * cp[row * 66 + col16];
      }
  }
  __syncthreads();

  // fused = h @ fw^T + fb   (K = 192)
  v8f facc[4];
#pragma unroll
  for (int c = 0; c < 4; ++c) {
    float bb = fb[16 * c + l16];
#pragma unroll
    for (int r = 0; r < 8; ++r) facc[c][r] = bb;
  }
#pragma unroll
  for (int j = 0; j < 48; ++j) {
    int kk = 4 * j + 2 * hl;
    v2f a = *(const v2f*)(sH + l16 * 196 + kk);
#pragma unroll
    for (int c = 0; c < 4; ++c) {
      v2f b2 = *(const v2f*)(fw + (size_t)(16 * c + l16) * 192 + kk);
      facc[c] = wmma4(a, b2, facc[c]);
    }
  }

  // out = sigmoid((fused + residual) . ow + ob)
  float ob0 = ob[0];
#pragma unroll
  for (int r = 0; r < 8; ++r) {
    float part = 0.0f;
#pragma unroll
    for (int c = 0; c < 4; ++c)
      part += ow[16 * c + l16] * (facc[c][r] + racc[c][r]);
    part += __shfl_xor(part, 8);
    part += __shfl_xor(part, 4);
    part += __shfl_xor(part, 2);
    part += __shfl_xor(part, 1);
    if (l16 == 0) out[rowb + r + 8 * hl] = sigm(part + ob0);
  }
}

// ---------------------------------------------------------------------------
extern "C" void kernel_launch(void* const* d_in, const int* in_sizes, int n_in,
                              void* d_out, int out_size, void* d_ws, size_t ws_size,
                              hipStream_t stream) {
  const float* self_x = (const float*)d_in[0];
  const float* conv_x = (const float*)d_in[1];
  const float* q1w = (const float*)d_in[2];  const float* q1b = (const float*)d_in[3];
  const float* k1w = (const float*)d_in[4];  const float* k1b = (const float*)d_in[5];
  const float* v1w = (const float*)d_in[6];  const float* v1b = (const float*)d_in[7];
  const float* q2w = (const float*)d_in[8];  const float* q2b = (const float*)d_in[9];
  const float* k2w = (const float*)d_in[10]; const float* k2b = (const float*)d_in[11];
  const float* v2w = (const float*)d_in[12]; const float* v2b = (const float*)d_in[13];
  const float* p1w = (const float*)d_in[14]; const float* p1b = (const float*)d_in[15];
  const float* p2w = (const float*)d_in[16]; const float* p2b = (const float*)d_in[17];
  const float* tw  = (const float*)d_in[18]; const float* tb  = (const float*)d_in[19];
  const float* gw  = (const float*)d_in[20]; const float* gb  = (const float*)d_in[21];
  const float* fw  = (const float*)d_in[22]; const float* fb  = (const float*)d_in[23];
  const float* rw  = (const float*)d_in[24]; const float* rb  = (const float*)d_in[25];
  const float* ow  = (const float*)d_in[26]; const float* ob  = (const float*)d_in[27];
  (void)in_sizes; (void)n_in; (void)out_size; (void)ws_size;

  float* ws = (float*)d_ws;  // needs 8 * 8192 * 64 floats = 16 MB
  float* out = (float*)d_out;

  proj_kernel<<<dim3(NROWS / 16, 6), 32, 0, stream>>>(
      self_x, conv_x, q1w, q1b, k1w, k1b, v1w, v1b, q2w, q2b, k2w, k2b, v2w, v2b, ws);

  flash_kernel<<<dim3(NROWS / 64, 2), 128, 0, stream>>>(ws);

  tail_kernel<<<dim3(NROWS / 16), 32, 0, stream>>>(
      ws + (size_t)6 * NFELEM, ws + (size_t)7 * NFELEM,
      p1w, p1b, p2w, p2b, tw, tb, gw, gb, fw, fb, rw, rb, ow, ob, out);
}